// Net2_1236950581662
// MI455X (gfx1250) — compile-verified
//
#include <hip/hip_runtime.h>
#include <hip/hip_bf16.h>

#define GCN_N_NODES 100000
#define GCN_N_EDGES 3200000

typedef float v2f __attribute__((ext_vector_type(2)));
typedef float v8f __attribute__((ext_vector_type(8)));

// ---------------------------------------------------------------------------
// Degree accumulation: one float atomic per endpoint per edge.
// ---------------------------------------------------------------------------
__global__ void gcn_degree_kernel(const int* __restrict__ src,
                                  const int* __restrict__ dst,
                                  float* __restrict__ deg_src,
                                  float* __restrict__ deg_dst,
                                  int n_edges) {
    int e = blockIdx.x * blockDim.x + threadIdx.x;
    if (e < n_edges) {
        atomicAdd(&deg_src[src[e]], 1.0f);
        atomicAdd(&deg_dst[dst[e]], 1.0f);
    }
}

// In-place: v = rsqrt(max(v, 1))
__global__ void gcn_norm_kernel(float* __restrict__ v, int n) {
    int i = blockIdx.x * blockDim.x + threadIdx.x;
    if (i < n) v[i] = rsqrtf(fmaxf(v[i], 1.0f));
}

// ---------------------------------------------------------------------------
// WMMA fp32 GEMM: Y[row, n] = sum_k (X[row,k]*scale[row]) * W[k,n]
// One wave32 per 16-row tile; N tile = 16 (cols >= ncol are masked).
// Uses V_WMMA_F32_16X16X4_F32 (exact fp32, matches reference precision).
//   A 16x4 layout: lanes 0-15 -> (M=lane, K+0 / K+1), lanes 16-31 -> (K+2 / K+3)
//   B 4x16 layout: mirrored; C/D: VGPR r -> M = r + 8*(lane>=16), N = lane&15
// ---------------------------------------------------------------------------
__global__ __launch_bounds__(256)
void gcn_gemm_wmma(const float* __restrict__ X, int ldx, int K,
                   const float* __restrict__ scale,
                   const float* __restrict__ W, int ncol,   // W is K x ncol, row-major
                   float* __restrict__ Y, int ldy, int nrows) {
    const int lane = threadIdx.x & 31;
    const int wave = threadIdx.x >> 5;
    const int tile = blockIdx.x * (blockDim.x >> 5) + wave;
    if (tile * 16 >= nrows) return;           // wave-uniform exit

    const int m    = lane & 15;               // M (and N for B/C) index
    const int half = lane >> 4;               // 0 -> K+0/K+1, 1 -> K+2/K+3

    int row = tile * 16 + m;
    if (row >= nrows) row = nrows - 1;        // clamp (nrows is a multiple of 16)
    const float s = scale[row];

    const float* xr = X + (size_t)row * ldx + 2 * half;
    const int ncl = (m < ncol) ? m : (ncol - 1);
    const float  bmask = (m < ncol) ? 1.0f : 0.0f;
    const float* wp = W + (size_t)(2 * half) * ncol + ncl;

    v8f c = {};
    const int kfull = K & ~3;

#pragma unroll 4
    for (int k = 0; k < kfull; k += 4) {
        v2f a, b;
        a.x = xr[k]     * s;
        a.y = xr[k + 1] * s;
        b.x = wp[(size_t)k       * ncol] * bmask;
        b.y = wp[(size_t)(k + 1) * ncol] * bmask;
        c = __builtin_amdgcn_wmma_f32_16x16x4_f32(
                /*neg_a=*/false, a, /*neg_b=*/false, b,
                /*c_mod=*/(short)0, c, /*reuse_a=*/false, /*reuse_b=*/false);
    }

    const int kr = K - kfull;                 // 0..3 tail K values
    if (kr) {
        float ax = 0.f, ay = 0.f, bx = 0.f, by = 0.f;
        if (2 * half < kr) {
            ax = xr[kfull] * s;
            bx = wp[(size_t)kfull * ncol] * bmask;
        }
        if (2 * half + 1 < kr) {
            ay = xr[kfull + 1] * s;
            by = wp[(size_t)(kfull + 1) * ncol] * bmask;
        }
        // EXEC reconverges before the WMMA (required: all lanes active)
        v2f a = {ax, ay}, b = {bx, by};
        c = __builtin_amdgcn_wmma_f32_16x16x4_f32(
                false, a, false, b, (short)0, c, false, false);
    }

    if (m < ncol) {
        float* yp = Y + (size_t)(tile * 16 + half * 8) * ldy + m;
#pragma unroll
        for (int r = 0; r < 8; ++r)
            yp[(size_t)r * ldy] = c[r];
    }
}

// ---------------------------------------------------------------------------
// Edge scatter-add: A[dst[e]*ld + j] += H[src[e]*ld + j]
// tid = e*ld + j -> 64B-coalesced gather and atomic per edge.
// ---------------------------------------------------------------------------
__global__ void gcn_scatter_kernel(const float* __restrict__ H,
                                   const int* __restrict__ src,
                                   const int* __restrict__ dst,
                                   float* __restrict__ A,
                                   int n_edges, int shift, int ncol) {
    unsigned t = blockIdx.x * blockDim.x + threadIdx.x;
    const int ld = 1 << shift;
    int j = t & (ld - 1);
    unsigned e = t >> shift;
    if (e < (unsigned)n_edges && j < ncol) {
        int s = src[e], d = dst[e];
        atomicAdd(&A[(size_t)d * ld + j], H[(size_t)s * ld + j]);
    }
}

// Hout = relu(A*norm_dst + b1), layout N x 16
__global__ void gcn_finish1_kernel(const float* __restrict__ A,
                                   const float* __restrict__ norm_dst,
                                   const float* __restrict__ b1,
                                   float* __restrict__ Hout, int n) {
    int i = blockIdx.x * blockDim.x + threadIdx.x;
    if (i < n) {
        int node = i >> 4, j = i & 15;
        float v = A[i] * norm_dst[node] + b1[j];
        Hout[i] = v > 0.f ? v : 0.f;
    }
}

// out = A2*norm_dst + b2, A2 layout N x 8 (7 valid), out layout N x 7
__global__ void gcn_finish2_kernel(const float* __restrict__ A2,
                                   const float* __restrict__ norm_dst,
                                   const float* __restrict__ b2,
                                   float* __restrict__ out, int n_nodes) {
    int i = blockIdx.x * blockDim.x + threadIdx.x;
    if (i < n_nodes * 7) {
        int node = i / 7;
        int j = i - node * 7;
        out[i] = A2[(size_t)node * 8 + j] * norm_dst[node] + b2[j];
    }
}

// ---------------------------------------------------------------------------
extern "C" void kernel_launch(void* const* d_in, const int* in_sizes, int n_in,
                              void* d_out, int out_size, void* d_ws, size_t ws_size,
                              hipStream_t stream) {
    (void)in_sizes; (void)n_in; (void)out_size; (void)ws_size;
    const float* features = (const float*)d_in[0];   // [N, 1433]
    const int*   src      = (const int*)  d_in[1];   // [E]
    const int*   dst      = (const int*)  d_in[2];   // [E]
    const float* W1       = (const float*)d_in[3];   // [1433, 16]
    const float* b1       = (const float*)d_in[4];   // [16]
    const float* W2       = (const float*)d_in[5];   // [16, 7]
    const float* b2       = (const float*)d_in[6];   // [7]
    float*       out      = (float*)d_out;           // [N, 7]

    const int N = GCN_N_NODES;
    const int E = GCN_N_EDGES;

    // Workspace layout (floats): zero-init region first (26N), then outputs.
    float* f        = (float*)d_ws;
    float* norm_src = f;                  // N   (deg_out -> rsqrt)
    float* norm_dst = f + N;              // N   (deg_in  -> rsqrt)
    float* A1       = f + 2 * (size_t)N;  // 16N (layer-1 aggregation, atomics)
    float* A2       = f + 18 * (size_t)N; // 8N  (layer-2 aggregation, atomics)
    float* H1       = f + 26 * (size_t)N; // 16N (GEMM1 out, then relu'd layer-1 out)
    float* H2       = f + 42 * (size_t)N; // 8N  (GEMM2 out, ld=8, 7 valid)

    // Zero the atomic-accumulated regions (every call: graph replay safe).
    hipMemsetAsync(f, 0, (size_t)26 * N * sizeof(float), stream);

    // Degrees + norms
    gcn_degree_kernel<<<(E + 255) / 256, 256, 0, stream>>>(src, dst, norm_src, norm_dst, E);
    gcn_norm_kernel<<<(2 * N + 255) / 256, 256, 0, stream>>>(f, 2 * N);

    // Layer 1: H1 = (X * norm_src) @ W1   (K=1433, N=16)
    {
        int tiles = (N + 15) / 16;                 // 6250
        int wpb   = 256 / 32;                      // 8 waves per block
        int grid  = (tiles + wpb - 1) / wpb;
        gcn_gemm_wmma<<<grid, 256, 0, stream>>>(features, 1433, 1433,
                                                norm_src, W1, 16, H1, 16, N);
    }
    // A1[dst] += H1[src]
    {
        unsigned total = (unsigned)E << 4;
        gcn_scatter_kernel<<<(total + 255) / 256, 256, 0, stream>>>(H1, src, dst, A1, E, 4, 16);
    }
    // H1 <- relu(A1 * norm_dst + b1)
    gcn_finish1_kernel<<<(16 * N + 255) / 256, 256, 0, stream>>>(A1, norm_dst, b1, H1, 16 * N);

    // Layer 2: H2 = (H1 * norm_src) @ W2   (K=16, N=7, padded to ld=8)
    {
        int tiles = (N + 15) / 16;
        int wpb   = 256 / 32;
        int grid  = (tiles + wpb - 1) / wpb;
        gcn_gemm_wmma<<<grid, 256, 0, stream>>>(H1, 16, 16,
                                                norm_src, W2, 7, H2, 8, N);
    }
    // A2[dst] += H2[src]
    {
        unsigned total = (unsigned)E << 3;
        gcn_scatter_kernel<<<(total + 255) / 256, 256, 0, stream>>>(H2, src, dst, A2, E, 3, 7);
    }
    // out = A2 * norm_dst + b2
    gcn_finish2_kernel<<<(7 * N + 255) / 256, 256, 0, stream>>>(A2, norm_dst, b2, out, N);
}